// GAT_31842887533271
// MI455X (gfx1250) — compile-verified
//
#include <hip/hip_runtime.h>
#include <hip/hip_bf16.h>
#include <math.h>

// Problem constants (from reference)
#define NNODES   50000
#define NEDGES   800000
#define INC      128
#define HIDC     64
#define NHEADS   4
#define NEG_SLOPE 0.2f
#define EPS_F    1e-16f

typedef __attribute__((ext_vector_type(2))) float v2f;
typedef __attribute__((ext_vector_type(8))) float v8f;

// ---------------------------------------------------------------------------
// fp32 WMMA GEMM: C[M,NC] = A[M,K] * B[K,NC], row-major, one wave per 16x16
// tile, K stepped by 4 via V_WMMA_F32_16X16X4_F32 (exact fp32 semantics).
// Requires M%16==0, NC%16==0, K%4==0.  EXEC all-ones at every WMMA: the only
// branch is wave-uniform.
// ---------------------------------------------------------------------------
__global__ void wmma_gemm_f32(const float* __restrict__ A,
                              const float* __restrict__ B,
                              float* __restrict__ C,
                              int M, int K, int NC) {
  const int lane = threadIdx.x & 31;
  const int wave = threadIdx.x >> 5;
  const int wavesPerBlock = blockDim.x >> 5;
  const int tile = blockIdx.x * wavesPerBlock + wave;
  const int tilesPerRow = NC >> 4;
  const int nTiles = (M >> 4) * tilesPerRow;
  if (tile >= nTiles) return;                  // wave-uniform guard
  const int rowBase = (tile / tilesPerRow) << 4;
  const int colBase = (tile % tilesPerRow) << 4;
  const int lo = lane & 15;                    // M index (A) / N index (B,C)
  const int hi = lane >> 4;                    // K-pair select (A,B), M+8 (C)

  v8f acc = {};
  const float* arow = A + (size_t)(rowBase + lo) * K;  // A: row lo, K striped
  const float* bcol = B + colBase + lo;                // B: col lo, K striped
  for (int k0 = 0; k0 < K; k0 += 4) {
    const int ka = k0 + 2 * hi;            // lanes 0-15: K=k0,k0+1 ; 16-31: +2
    v2f a; a[0] = arow[ka];
           a[1] = arow[ka + 1];
    v2f b; b[0] = bcol[(size_t)ka * NC];
           b[1] = bcol[(size_t)(ka + 1) * NC];
    acc = __builtin_amdgcn_wmma_f32_16x16x4_f32(
        /*neg_a=*/false, a, /*neg_b=*/false, b,
        /*c_mod=*/(short)0, acc, /*reuse_a=*/false, /*reuse_b=*/false);
  }
  // C layout: VGPR r -> row (rowBase + hi*8 + r), col (colBase + lo)
  float* crow = C + (size_t)(rowBase + hi * 8) * NC + colBase + lo;
#pragma unroll
  for (int r = 0; r < 8; ++r) crow[(size_t)r * NC] = acc[r];
}

// ---------------------------------------------------------------------------
// Per-node attention coefficients: as[n,h] = <h[n,h,:], a_src[h,:]>, same dst.
// ---------------------------------------------------------------------------
__global__ void attn_coeff(const float* __restrict__ h,
                           const float* __restrict__ a_src,
                           const float* __restrict__ a_dst,
                           float* __restrict__ as_, float* __restrict__ ad_,
                           int n, int H, int C) {
  const int node = blockIdx.x * blockDim.x + threadIdx.x;
  if (node >= n) return;
  const float* row = h + (size_t)node * H * C;
  for (int hh = 0; hh < H; ++hh) {
    float ss = 0.f, dd = 0.f;
    for (int c = 0; c < C; ++c) {
      const float v = row[hh * C + c];
      ss = fmaf(v, a_src[hh * C + c], ss);
      dd = fmaf(v, a_dst[hh * C + c], dd);
    }
    as_[node * H + hh] = ss;
    ad_[node * H + hh] = dd;
  }
}

__global__ void init_softmax(float* __restrict__ m, float* __restrict__ dn,
                             int n) {
  const int i = blockIdx.x * blockDim.x + threadIdx.x;
  if (i < n) { m[i] = -__builtin_inff(); dn[i] = 0.f; }
}

__global__ void init_out_bias(float* __restrict__ out,
                              const float* __restrict__ bias, int n, int Ct) {
  const int i = blockIdx.x * blockDim.x + threadIdx.x;
  if (i < n * Ct) out[i] = bias[i % Ct];
}

// sign-magnitude trick: works for all float orderings incl. -inf init
__device__ __forceinline__ void atomicMaxFloat(float* addr, float v) {
  if (v >= 0.f)
    atomicMax((int*)addr, __float_as_int(v));
  else
    atomicMin((unsigned int*)addr, __float_as_uint(v));
}

__device__ __forceinline__ float lrelu(float x) {
  return x >= 0.f ? x : NEG_SLOPE * x;
}

// Pass 1: segment max over incoming edges of each dst node
__global__ void edge_max(const long long* __restrict__ ei,
                         const float* __restrict__ as_,
                         const float* __restrict__ ad_,
                         float* __restrict__ m, int H) {
  const int t = blockIdx.x * blockDim.x + threadIdx.x;
  if (t >= NEDGES * H) return;
  const int e = t / H, hh = t - e * H;
  const int s = (int)ei[e];
  const int d = (int)ei[NEDGES + e];
  atomicMaxFloat(&m[d * H + hh], lrelu(as_[s * H + hh] + ad_[d * H + hh]));
}

// Pass 2: ex = exp(e - m[dst]); denom += ex; stash ex in alpha buffer
__global__ void edge_expsum(const long long* __restrict__ ei,
                            const float* __restrict__ as_,
                            const float* __restrict__ ad_,
                            const float* __restrict__ m,
                            float* __restrict__ dn,
                            float* __restrict__ alpha, int H) {
  const int t = blockIdx.x * blockDim.x + threadIdx.x;
  if (t >= NEDGES * H) return;
  const int e = t / H, hh = t - e * H;
  const int s = (int)ei[e];
  const int d = (int)ei[NEDGES + e];
  const float v = lrelu(as_[s * H + hh] + ad_[d * H + hh]);
  const float ex = expf(v - m[d * H + hh]);
  atomicAdd(&dn[d * H + hh], ex);
  alpha[t] = ex;
}

// Pass 3: alpha /= (denom[dst] + eps)
__global__ void edge_norm(const long long* __restrict__ ei,
                          const float* __restrict__ dn,
                          float* __restrict__ alpha, int H) {
  const int t = blockIdx.x * blockDim.x + threadIdx.x;
  if (t >= NEDGES * H) return;
  const int e = t / H, hh = t - e * H;
  const int d = (int)ei[NEDGES + e];
  alpha[t] = alpha[t] / (dn[d * H + hh] + EPS_F);
}

// Pass 4: out[dst,h,c] += h[src,h,c] * alpha[e,h]  (64 threads per edge)
__global__ void edge_scatter(const long long* __restrict__ ei,
                             const float* __restrict__ hsrc,
                             const float* __restrict__ alpha,
                             float* __restrict__ out, int H, int C) {
  const int t = blockIdx.x * blockDim.x + threadIdx.x;
  if (t >= NEDGES * C) return;
  const int e = t / C, c = t - e * C;
  const int s = (int)ei[e];
  const int d = (int)ei[NEDGES + e];
  for (int hh = 0; hh < H; ++hh) {
    const float av = alpha[(size_t)e * H + hh];
    atomicAdd(&out[(size_t)d * H * C + hh * C + c],
              hsrc[(size_t)s * H * C + hh * C + c] * av);
  }
}

__global__ void elu_inplace(float* __restrict__ x, int n) {
  const int i = blockIdx.x * blockDim.x + threadIdx.x;
  if (i < n) {
    const float v = x[i];
    x[i] = v > 0.f ? v : expm1f(v);
  }
}

// ---------------------------------------------------------------------------
extern "C" void kernel_launch(void* const* d_in, const int* in_sizes, int n_in,
                              void* d_out, int out_size, void* d_ws, size_t ws_size,
                              hipStream_t stream) {
  const float*     x      = (const float*)d_in[0];
  const long long* ei     = (const long long*)d_in[1];  // int64 [2,E]
  const float*     W1     = (const float*)d_in[2];
  const float*     a_src1 = (const float*)d_in[3];
  const float*     a_dst1 = (const float*)d_in[4];
  const float*     b1     = (const float*)d_in[5];
  const float*     W2     = (const float*)d_in[6];
  const float*     a_src2 = (const float*)d_in[7];
  const float*     a_dst2 = (const float*)d_in[8];
  const float*     b2     = (const float*)d_in[9];
  float* out = (float*)d_out;

  const int N = NNODES, E = NEDGES;
  const int C1 = NHEADS * HIDC;  // 256

  // workspace carve-up (~135 MB)
  float* ws  = (float*)d_ws;
  float* h1  = ws;                         // N*256
  float* o1  = h1  + (size_t)N * C1;       // N*256 (accum -> ELU -> layer2 in)
  float* h2  = o1  + (size_t)N * C1;       // N*64
  float* as1 = h2  + (size_t)N * HIDC;     // N*4
  float* ad1 = as1 + (size_t)N * NHEADS;
  float* m1  = ad1 + (size_t)N * NHEADS;
  float* dn1 = m1  + (size_t)N * NHEADS;
  float* as2 = dn1 + (size_t)N * NHEADS;   // N
  float* ad2 = as2 + (size_t)N;
  float* m2  = ad2 + (size_t)N;
  float* dn2 = m2  + (size_t)N;
  float* al1 = dn2 + (size_t)N;            // E*4
  float* al2 = al1 + (size_t)E * NHEADS;   // E

  const int T = 256;
  auto cdiv = [](long long a, long long b) { return (int)((a + b - 1) / b); };

  // ---------------- layer 1 ----------------
  {
    const int tiles = (N / 16) * (C1 / 16);              // 50000
    wmma_gemm_f32<<<tiles / 4, 128, 0, stream>>>(x, W1, h1, N, INC, C1);
  }
  attn_coeff<<<cdiv(N, T), T, 0, stream>>>(h1, a_src1, a_dst1, as1, ad1,
                                           N, NHEADS, HIDC);
  init_softmax<<<cdiv((long long)N * NHEADS, T), T, 0, stream>>>(m1, dn1, N * NHEADS);
  init_out_bias<<<cdiv((long long)N * C1, T), T, 0, stream>>>(o1, b1, N, C1);
  edge_max   <<<cdiv((long long)E * NHEADS, T), T, 0, stream>>>(ei, as1, ad1, m1, NHEADS);
  edge_expsum<<<cdiv((long long)E * NHEADS, T), T, 0, stream>>>(ei, as1, ad1, m1, dn1, al1, NHEADS);
  edge_norm  <<<cdiv((long long)E * NHEADS, T), T, 0, stream>>>(ei, dn1, al1, NHEADS);
  edge_scatter<<<cdiv((long long)E * HIDC, T), T, 0, stream>>>(ei, h1, al1, o1, NHEADS, HIDC);
  elu_inplace<<<cdiv((long long)N * C1, T), T, 0, stream>>>(o1, N * C1);

  // ---------------- layer 2 ----------------
  {
    const int tiles = (N / 16) * (HIDC / 16);            // 12500
    wmma_gemm_f32<<<tiles / 4, 128, 0, stream>>>(o1, W2, h2, N, C1, HIDC);
  }
  attn_coeff<<<cdiv(N, T), T, 0, stream>>>(h2, a_src2, a_dst2, as2, ad2,
                                           N, 1, HIDC);
  init_softmax<<<cdiv(N, T), T, 0, stream>>>(m2, dn2, N);
  init_out_bias<<<cdiv((long long)N * HIDC, T), T, 0, stream>>>(out, b2, N, HIDC);
  edge_max   <<<cdiv(E, T), T, 0, stream>>>(ei, as2, ad2, m2, 1);
  edge_expsum<<<cdiv(E, T), T, 0, stream>>>(ei, as2, ad2, m2, dn2, al2, 1);
  edge_norm  <<<cdiv(E, T), T, 0, stream>>>(ei, dn2, al2, 1);
  edge_scatter<<<cdiv((long long)E * HIDC, T), T, 0, stream>>>(ei, h2, al2, out, 1, HIDC);
  elu_inplace<<<cdiv((long long)N * HIDC, T), T, 0, stream>>>(out, N * HIDC);
}